// SCDM_Attention_14998025798075
// MI455X (gfx1250) — compile-verified
//
#include <hip/hip_runtime.h>
#include <hip/hip_bf16.h>

typedef float v2f __attribute__((ext_vector_type(2)));
typedef float v8f __attribute__((ext_vector_type(8)));

#define WMMA_F32(a, b, c) \
    __builtin_amdgcn_wmma_f32_16x16x4_f32(false, (a), false, (b), (short)0, (c), false, false)

#if defined(__has_builtin) && __has_builtin(__builtin_amdgcn_sched_barrier)
#define SCHED_FENCE() __builtin_amdgcn_sched_barrier(0)
#else
#define SCHED_FENCE()
#endif

// XOR-swizzled LDS pair index (8-byte granularity): conflict-free for both the
// cooperative stores (rows split by parity into disjoint bank halves) and the
// WMMA fragment reads (p vs p+1 differ in bit0 -> lane groups cannot collide).
__device__ __forceinline__ int sw(int row, int p) {
    return (row << 4) | (p ^ (row & 15));
}

// One BK=32 chunk: 8 k-steps x 4 WMMAs. Fragment loads for step s+1 are issued
// before step s's WMMAs and PINNED there with sched_barrier(0) so the machine
// scheduler cannot re-sink them onto their consumers; the wait in front of each
// WMMA group then covers only the 3 newer in-flight ds_loads (dscnt<=3), whose
// latency is hidden under the previous WMMA group.
__device__ __forceinline__ void compute_chunk(
    const v2f* __restrict__ bA, const v2f* __restrict__ bB,
    int aRow, int ml, int pg0, v8f acc[4])
{
    v2f a_c  = bA[sw(aRow, pg0)];
    v2f b0_c = bB[sw(ml, pg0)];
    v2f b1_c = bB[sw(16 + ml, pg0)];
    v2f b2_c = bB[sw(32 + ml, pg0)];
    v2f b3_c = bB[sw(48 + ml, pg0)];

    #pragma unroll
    for (int s = 0; s < 8; ++s) {
        v2f a_n, b0_n, b1_n, b2_n, b3_n;
        if (s < 7) {
            const int pp = 2 * (s + 1) + pg0;
            a_n  = bA[sw(aRow, pp)];
            b0_n = bB[sw(ml, pp)];
            b1_n = bB[sw(16 + ml, pp)];
            b2_n = bB[sw(32 + ml, pp)];
            b3_n = bB[sw(48 + ml, pp)];
        }
        SCHED_FENCE();                       // keep s+1 loads above s WMMAs
        acc[0] = WMMA_F32(a_c, b0_c, acc[0]);
        acc[1] = WMMA_F32(a_c, b1_c, acc[1]);
        acc[2] = WMMA_F32(a_c, b2_c, acc[2]);
        acc[3] = WMMA_F32(a_c, b3_c, acc[3]);
        SCHED_FENCE();                       // keep s WMMAs above s+2 loads
        if (s < 7) { a_c = a_n; b0_c = b0_n; b1_c = b1_n; b2_c = b2_n; b3_c = b3_n; }
    }
}

// ---------------------------------------------------------------------------
// GEMM: C[m,h] = sum_k A[m,k] * W[h,k] (+ bias[h])   (A:[M,K], W:[H,K] row-major)
// Block = 128 threads (4 waves): 64(M) x 64(H) tile, BK = 32 per chunk.
// A/B chunks staged through double-buffered, XOR-swizzled LDS (32 KB total);
// chunk c+1 global loads (coalesced b64) prefetch into registers while chunk
// c's 32 WMMAs consume conflict-free, register-pipelined ds_load fragments.
// Fragment layout per CDNA5 ISA 7.12.2:
//   A 16x4 f32: lane holds row M=lane%16; VGPR i holds K = k0 + 2*(lane/16) + i
//   B 4x16 f32: lane holds col N=lane%16; VGPR i holds K = k0 + 2*(lane/16) + i
//   C/D 16x16:  lane holds col N=lane%16; VGPR r holds row M = r + 8*(lane/16)
// ---------------------------------------------------------------------------
__global__ __launch_bounds__(128) void scdm_gemm_wmma_f32(
    const float* __restrict__ A, const float* __restrict__ W,
    const float* __restrict__ bias, float* __restrict__ C,
    int M, int K, int H)
{
    __shared__ v2f tA[2][64 * 16];   // [buf][row*16 + swizzled pair]  (8 KB each)
    __shared__ v2f tB[2][64 * 16];

    const int tid  = threadIdx.x;
    const int lane = tid & 31;
    const int wid  = tid >> 5;
    const int m0   = blockIdx.x * 64;          // block M base
    const int h0   = blockIdx.y * 64;          // block H base
    const int ml   = lane & 15;
    const int pg0  = lane >> 4;                // pair sub-index (0/1) per lane group
    const int aRow = wid * 16 + ml;            // block-local A row for this lane

    // cooperative-load mapping: thread owns pair column p, rows r0 + 8j
    const int p  = tid & 15;
    const int r0 = tid >> 4;
    const float* Ablk = A + (size_t)m0 * K + 2 * p;
    const float* Wblk = W + (size_t)h0 * K + 2 * p;

    v8f acc[4] = {};
    v2f preA[8], preB[8];

    const int C32 = K >> 5;                    // number of BK=32 chunks

    // prologue: chunk 0 -> buffer 0
    #pragma unroll
    for (int j = 0; j < 8; ++j) {
        preA[j] = *(const v2f*)(Ablk + (size_t)(r0 + 8 * j) * K);
        preB[j] = *(const v2f*)(Wblk + (size_t)(r0 + 8 * j) * K);
    }
    #pragma unroll
    for (int j = 0; j < 8; ++j) {
        tA[0][sw(r0 + 8 * j, p)] = preA[j];
        tB[0][sw(r0 + 8 * j, p)] = preB[j];
    }
    __syncthreads();

    #pragma unroll 1
    for (int c = 0; c < C32 - 1; ++c) {
        const int k0 = (c + 1) << 5;
        // prefetch chunk c+1 (coalesced; overlaps WMMAs below)
        #pragma unroll
        for (int j = 0; j < 8; ++j) {
            preA[j] = *(const v2f*)(Ablk + (size_t)(r0 + 8 * j) * K + k0);
            preB[j] = *(const v2f*)(Wblk + (size_t)(r0 + 8 * j) * K + k0);
        }
        // compute chunk c from LDS buffer c&1
        compute_chunk(tA[c & 1], tB[c & 1], aRow, ml, pg0, acc);

        // store chunk c+1 into the other buffer (disjoint from readers of c&1)
        v2f* dA = tA[(c + 1) & 1];
        v2f* dB = tB[(c + 1) & 1];
        #pragma unroll
        for (int j = 0; j < 8; ++j) {
            dA[sw(r0 + 8 * j, p)] = preA[j];
            dB[sw(r0 + 8 * j, p)] = preB[j];
        }
        __syncthreads();
    }

    // epilogue chunk
    compute_chunk(tA[(C32 - 1) & 1], tB[(C32 - 1) & 1], aRow, ml, pg0, acc);

    const int mb = m0 + wid * 16 + ((lane >> 4) << 3);
    #pragma unroll
    for (int t = 0; t < 4; ++t) {
        const int h  = h0 + 16 * t + ml;
        const float bv = bias ? bias[h] : 0.0f;
        #pragma unroll
        for (int r = 0; r < 8; ++r)
            C[(size_t)(mb + r) * H + h] = acc[t][r] + bv;
    }
}

// ---------------------------------------------------------------------------
// Fused scores + softmax + context.
// Block = 256 threads (8 waves), one wave per t; grid = (T/8, B).
// LDS (64 KB) stages s_proj[b] in two 512-wide H halves ([n][h] layout ->
// conflict-free reads: lane = h%32), then is reused for sent_feat[b].
// Scores held lane-diagonally (lane n owns score_n) for wave32 softmax.
// ---------------------------------------------------------------------------
__global__ __launch_bounds__(256) void scdm_attn_fused(
    const float* __restrict__ v_proj,   // [B*T, 1024]
    const float* __restrict__ s_proj,   // [B*N, 1024]
    const float* __restrict__ sent,     // [B, N, 512]
    const float* __restrict__ wv,       // [1024]
    float* __restrict__ out)            // [B*T, 512]
{
    __shared__ float lds[32 * 512];     // 64 KB

    const int b    = blockIdx.y;
    const int lane = threadIdx.x & 31;
    const int wid  = threadIdx.x >> 5;
    const int t    = blockIdx.x * 8 + wid;

    float score = 0.0f;                 // lane n holds score for sentence n

    for (int half = 0; half < 2; ++half) {
        __syncthreads();                // previous LDS consumers done
        // cooperative load s_proj[b, n, half*512 + h] -> lds[n*512 + h]
        for (int i = threadIdx.x; i < 32 * 512; i += 256) {
            const int n = i >> 9, h = i & 511;
            lds[i] = s_proj[(size_t)(b * 32 + n) * 1024 + half * 512 + h];
        }
        __syncthreads();

        // per-lane register cache of v_proj row chunk and wv chunk
        float v[16], w[16];
        const float* vp = v_proj + (size_t)(b * 256 + t) * 1024 + half * 512;
        const float* wp = wv + half * 512;
        #pragma unroll
        for (int j = 0; j < 16; ++j) { v[j] = vp[lane + 32 * j]; w[j] = wp[lane + 32 * j]; }

        for (int n = 0; n < 32; ++n) {
            const float* srow = lds + n * 512;
            float pacc = 0.0f;
            #pragma unroll
            for (int j = 0; j < 16; ++j) {
                float x = v[j] + srow[lane + 32 * j];
                float th;
                // gfx1250 hardware tanh; v_nop satisfies TRANS-op result hazard
                asm("v_tanh_f32 %0, %1\n\tv_nop" : "=v"(th) : "v"(x));
                pacc = fmaf(w[j], th, pacc);
            }
            #pragma unroll
            for (int off = 16; off; off >>= 1) pacc += __shfl_xor(pacc, off, 32);
            if (lane == n) score += pacc;
        }
    }

    // softmax over N=32 (one value per lane)
    float mx = score;
    #pragma unroll
    for (int off = 16; off; off >>= 1) mx = fmaxf(mx, __shfl_xor(mx, off, 32));
    float e = __expf(score - mx);
    float sm = e;
    #pragma unroll
    for (int off = 16; off; off >>= 1) sm += __shfl_xor(sm, off, 32);
    const float attn = e / sm;

    // context: C[b,t,d] = sum_n attn[n] * sent[b,n,d]
    __syncthreads();
    for (int i = threadIdx.x; i < 32 * 512; i += 256)
        lds[i] = sent[(size_t)b * 32 * 512 + i];
    __syncthreads();

    float acc[16];
    #pragma unroll
    for (int j = 0; j < 16; ++j) acc[j] = 0.0f;
    for (int n = 0; n < 32; ++n) {
        const float a = __shfl(attn, n, 32);
        const float* srow = lds + n * 512;
        #pragma unroll
        for (int j = 0; j < 16; ++j) acc[j] = fmaf(a, srow[lane + 32 * j], acc[j]);
    }
    float* op = out + (size_t)(b * 256 + t) * 512;
    #pragma unroll
    for (int j = 0; j < 16; ++j) op[lane + 32 * j] = acc[j];
}

extern "C" void kernel_launch(void* const* d_in, const int* in_sizes, int n_in,
                              void* d_out, int out_size, void* d_ws, size_t ws_size,
                              hipStream_t stream) {
    (void)in_sizes; (void)n_in; (void)out_size; (void)ws_size;

    const float* video = (const float*)d_in[0];   // [16,256,1024]
    const float* sent  = (const float*)d_in[1];   // [16,32,512]
    const float* Ws    = (const float*)d_in[2];   // [1024,512]
    const float* Wa    = (const float*)d_in[3];   // [1024,1024]
    const float* ba    = (const float*)d_in[4];   // [1024]
    const float* wv    = (const float*)d_in[5];   // [1024]
    float* out = (float*)d_out;                   // [16,256,512]

    float* s_proj = (float*)d_ws;                 // 512*1024 f32 = 2 MB
    float* v_proj = s_proj + (size_t)512 * 1024;  // 4096*1024 f32 = 16.8 MB

    // s_proj[b,n,:] = sent @ Ws^T            (M=512, K=512, H=1024)
    scdm_gemm_wmma_f32<<<dim3(512 / 64, 1024 / 64), 128, 0, stream>>>(
        sent, Ws, nullptr, s_proj, 512, 512, 1024);

    // v_proj[b,t,:] = video @ Wa^T + ba      (M=4096, K=1024, H=1024)
    scdm_gemm_wmma_f32<<<dim3(4096 / 64, 1024 / 64), 128, 0, stream>>>(
        video, Wa, ba, v_proj, 4096, 1024, 1024);

    // fused tanh-scores + softmax + context
    scdm_attn_fused<<<dim3(256 / 8, 16), 256, 0, stream>>>(
        v_proj, s_proj, sent, wv, out);
}